// SModule_83399674954366
// MI455X (gfx1250) — compile-verified
//
#include <hip/hip_runtime.h>

typedef __attribute__((ext_vector_type(2))) float v2f;
typedef __attribute__((ext_vector_type(8))) float v8f;

// Problem constants (match reference)
constexpr int Bc = 2;
constexpr int Sc = 1024;
constexpr int Nc = Sc + 1;      // 1025 nodes (anchor + tokens)
constexpr int Dc = 512;
constexpr int Rc = 32;
constexpr int Lc = 3;
constexpr int Wc = 64;          // radius
constexpr int WNc = 2 * Wc + 1; // 129 neighbors
constexpr int WINc = 144;       // band rows touched by a 16-node tile: 16 + 2W
constexpr int NT = (Nc + 15) / 16; // 65 node tiles

// LDS strides (padded to avoid 64-bank conflicts on fragment reads)
constexpr int XSTR = Dc + 4;  // 516 : 4L mod 64 distinct over 16 lanes
constexpr int QSTR = 33;      // 33L mod 64 bijective
constexpr int KSTR = 33;
constexpr int SSTR = 146;
constexpr int WSTR = 145;     // 17L mod 64 bijective

__device__ __forceinline__ v8f wmma_f32(v2f a, v2f b, v8f c) {
  return __builtin_amdgcn_wmma_f32_16x16x4_f32(false, a, false, b, (short)0, c,
                                               false, false);
}

// ---------------------------------------------------------------------------
// Kernel 1: token embed + pos + input LayerNorm + state projection; anchor row
// grid: B*N blocks of 256 threads (each thread owns 2 of the 512 channels)
// ---------------------------------------------------------------------------
__device__ __forceinline__ float block_sum256(float v, float* red) {
  for (int m = 16; m >= 1; m >>= 1) v += __shfl_xor(v, m, 32);
  const int wave = threadIdx.x >> 5, lane = threadIdx.x & 31;
  __syncthreads();               // protect red[] reuse across calls
  if (lane == 0) red[wave] = v;
  __syncthreads();
  if (threadIdx.x < 8) {
    float t = red[threadIdx.x];
    t += __shfl_xor(t, 1, 32);
    t += __shfl_xor(t, 2, 32);
    t += __shfl_xor(t, 4, 32);
    if (threadIdx.x == 0) red[0] = t;
  }
  __syncthreads();
  return red[0];
}

__global__ __launch_bounds__(256) void embed_ln_kernel(
    const int* __restrict__ ids, const float* __restrict__ emb,
    const float* __restrict__ pos, const float* __restrict__ anchor_val,
    const float* __restrict__ anchor_state, const float* __restrict__ gamma,
    const float* __restrict__ beta, const float* __restrict__ sp_w,
    const float* __restrict__ sp_b, float* __restrict__ val,
    float* __restrict__ st) {
  __shared__ float red[8];
  const int n = blockIdx.x % Nc;
  const int b = blockIdx.x / Nc;
  const int tid = threadIdx.x;
  float* vrow = val + ((size_t)b * Nc + n) * Dc;
  if (n == 0) {  // anchor node
    vrow[tid] = anchor_val[tid];
    vrow[tid + 256] = anchor_val[tid + 256];
    if (tid == 0) st[(size_t)b * Nc] = anchor_state[0];
    return;
  }
  const int s = n - 1;
  const int id = ids[b * Sc + s];
  const float e0 = emb[(size_t)id * Dc + tid] + pos[(size_t)s * Dc + tid];
  const float e1 = emb[(size_t)id * Dc + tid + 256] + pos[(size_t)s * Dc + tid + 256];
  const float sum = block_sum256(e0 + e1, red);
  const float ssq = block_sum256(e0 * e0 + e1 * e1, red);
  const float mean = sum * (1.0f / Dc);
  const float var = ssq * (1.0f / Dc) - mean * mean;
  const float rs = rsqrtf(var + 1e-5f);
  const float y0 = (e0 - mean) * rs * gamma[tid] + beta[tid];
  const float y1 = (e1 - mean) * rs * gamma[tid + 256] + beta[tid + 256];
  vrow[tid] = y0;
  vrow[tid + 256] = y1;
  const float dot = block_sum256(y0 * sp_w[tid] + y1 * sp_w[tid + 256], red);
  if (tid == 0) st[(size_t)b * Nc + n] = dot + sp_b[0];
}

// ---------------------------------------------------------------------------
// Kernel 2 (per layer): x = LN(val); q = x@U; k = x@V  via fp32 WMMA
// grid: B*NT blocks of 128 threads (4 waves). Wave w -> {q,k} x {cols 0-15,16-31}
// Fast (unguarded) path for the 64/65 full tiles; guarded path for the tail.
// ---------------------------------------------------------------------------
__global__ __launch_bounds__(128) void ln_proj_kernel(
    const float* __restrict__ val, const float* __restrict__ gamma,
    const float* __restrict__ beta, const float* __restrict__ Uw,
    const float* __restrict__ Vw, float* __restrict__ xo,
    float* __restrict__ qo, float* __restrict__ ko) {
  __shared__ float xs[16 * XSTR];
  const int tile = blockIdx.x % NT;
  const int b = blockIdx.x / NT;
  const int n0 = tile * 16;
  const int tid = threadIdx.x;
  const bool full = (n0 + 16 <= Nc);  // wave-uniform

  // Each row handled by 8 threads, 64 channels each.
  const int row = tid >> 3, sub = tid & 7;
  const int g = n0 + row;
  const float* vrow = val + ((size_t)b * Nc + g) * Dc;
  if (full) {
    for (int c = sub * 64; c < sub * 64 + 64; ++c)
      xs[row * XSTR + c] = vrow[c];
  } else {
    for (int c = sub * 64; c < sub * 64 + 64; ++c)
      xs[row * XSTR + c] = (g < Nc) ? vrow[c] : 0.0f;
  }
  float s1 = 0.0f, s2 = 0.0f;
  for (int c = sub * 64; c < sub * 64 + 64; ++c) {
    const float v = xs[row * XSTR + c];
    s1 += v;
    s2 += v * v;
  }
  for (int m = 1; m < 8; m <<= 1) {
    s1 += __shfl_xor(s1, m, 32);
    s2 += __shfl_xor(s2, m, 32);
  }
  const float mean = s1 * (1.0f / Dc);
  const float var = s2 * (1.0f / Dc) - mean * mean;
  const float rs = rsqrtf(var + 1e-5f);
  float* xrow = xo + ((size_t)b * Nc + g) * Dc;
  if (full) {
    for (int c = sub * 64; c < sub * 64 + 64; ++c) {
      const float y = (xs[row * XSTR + c] - mean) * rs * gamma[c] + beta[c];
      xs[row * XSTR + c] = y;
      xrow[c] = y;
    }
  } else {
    for (int c = sub * 64; c < sub * 64 + 64; ++c) {
      const float y = (xs[row * XSTR + c] - mean) * rs * gamma[c] + beta[c];
      xs[row * XSTR + c] = y;
      if (g < Nc) xrow[c] = y;
    }
  }
  __syncthreads();

  // WMMA projection: [16 x 512] @ [512 x 16] per wave, K in steps of 4.
  const int wave = tid >> 5, lane = tid & 31;
  const int hf = lane >> 4, l16 = lane & 15;
  const float* Wm = (wave < 2) ? Uw : Vw;
  float* out = (wave < 2) ? qo : ko;
  const int cb = (wave & 1) * 16;
  v8f acc = {0.f, 0.f, 0.f, 0.f, 0.f, 0.f, 0.f, 0.f};
  for (int kk = 0; kk < Dc / 4; ++kk) {
    const int c0 = 4 * kk + 2 * hf;
    v2f a;
    a.x = xs[l16 * XSTR + c0];
    a.y = xs[l16 * XSTR + c0 + 1];
    v2f bv;
    bv.x = Wm[(size_t)c0 * Rc + cb + l16];
    bv.y = Wm[(size_t)(c0 + 1) * Rc + cb + l16];
    acc = wmma_f32(a, bv, acc);
  }
  if (full) {
    for (int i = 0; i < 8; ++i) {
      const int r = i + 8 * hf;
      out[((size_t)b * Nc + n0 + r) * Rc + cb + l16] = acc[i];
    }
  } else {
    for (int i = 0; i < 8; ++i) {
      const int r = i + 8 * hf, gg = n0 + r;
      if (gg < Nc) out[((size_t)b * Nc + gg) * Rc + cb + l16] = acc[i];
    }
  }
}

// ---------------------------------------------------------------------------
// Kernel 3 (per layer): banded scores (WMMA) -> signed-abs softmax -> banded
// delta GEMM (WMMA) -> residual add into val.
// grid: B*NT blocks of 128 threads. Interior tiles (56/65) take unguarded paths.
// ---------------------------------------------------------------------------
__global__ __launch_bounds__(128) void attn_kernel(
    const float* __restrict__ x, const float* __restrict__ q,
    const float* __restrict__ k, const float* __restrict__ st,
    float* __restrict__ val) {
  __shared__ float qs[16 * QSTR];
  __shared__ float ks[WINc * KSTR];
  __shared__ float ss[16 * SSTR];
  __shared__ float wm[16 * WSTR];
  __shared__ float sw[WINc];
  const int tile = blockIdx.x % NT;
  const int b = blockIdx.x / NT;
  const int n0 = tile * 16;
  const int tid = threadIdx.x;
  const bool full = (n0 + 16 <= Nc);                          // uniform
  const bool interior = (n0 >= Wc) && (n0 + 16 + Wc <= Nc);   // uniform

  // Stage q tile, k window, state window into LDS; zero the banded W matrix.
  const float* qb = q + ((size_t)b * Nc + n0) * Rc;
  const float* kb = k + ((size_t)b * Nc + n0 - Wc) * Rc;  // window base row
  const float* sb = st + (size_t)b * Nc + n0 - Wc;
  if (interior) {
    for (int i = tid; i < 16 * Rc; i += 128) {
      const int r = i >> 5, c = i & 31;
      qs[r * QSTR + c] = qb[r * Rc + c];
    }
    for (int i = tid; i < WINc * Rc; i += 128) {
      const int j = i >> 5, c = i & 31;
      ks[j * KSTR + c] = kb[(size_t)j * Rc + c];
    }
    for (int j = tid; j < WINc; j += 128) sw[j] = sb[j];
  } else {
    for (int i = tid; i < 16 * Rc; i += 128) {
      const int r = i >> 5, c = i & 31, g = n0 + r;
      qs[r * QSTR + c] = (g < Nc) ? q[((size_t)b * Nc + g) * Rc + c] : 0.0f;
    }
    for (int i = tid; i < WINc * Rc; i += 128) {
      const int j = i >> 5, c = i & 31, gn = n0 - Wc + j;
      ks[j * KSTR + c] =
          (gn >= 0 && gn < Nc) ? k[((size_t)b * Nc + gn) * Rc + c] : 0.0f;
    }
    for (int j = tid; j < WINc; j += 128) {
      const int gn = n0 - Wc + j;
      sw[j] = (gn >= 0 && gn < Nc) ? st[(size_t)b * Nc + gn] : 0.0f;
    }
  }
  for (int i = tid; i < 16 * WSTR; i += 128) wm[i] = 0.0f;
  __syncthreads();

  const int wave = tid >> 5, lane = tid & 31;
  const int hf = lane >> 4, l16 = lane & 15;

  // Scores S[16,144] = Q[16,32] * Kwin^T : 9 col tiles x 8 K-steps.
  for (int t = wave; t < 9; t += 4) {
    const int cb = t * 16;
    v8f acc = {0.f, 0.f, 0.f, 0.f, 0.f, 0.f, 0.f, 0.f};
    for (int kk = 0; kk < Rc / 4; ++kk) {
      const int c0 = 4 * kk + 2 * hf;
      v2f a;
      a.x = qs[l16 * QSTR + c0];
      a.y = qs[l16 * QSTR + c0 + 1];
      v2f bv;
      bv.x = ks[(cb + l16) * KSTR + c0];
      bv.y = ks[(cb + l16) * KSTR + c0 + 1];
      acc = wmma_f32(a, bv, acc);
    }
    for (int i = 0; i < 8; ++i) ss[(i + 8 * hf) * SSTR + cb + l16] = acc[i];
  }
  __syncthreads();

  // Masked signed-abs softmax, scaled by neighbor state -> banded Wmat.
  {
    const int row = tid >> 3, sub = tid & 7;
    const int g = n0 + row;
    if (interior) {
      float m = -1e30f;
      for (int o = sub; o < WNc; o += 8)
        m = fmaxf(m, fabsf(ss[row * SSTR + row + o]));
      for (int mm = 1; mm < 8; mm <<= 1) m = fmaxf(m, __shfl_xor(m, mm, 32));
      float Z = 0.0f;
      for (int o = sub; o < WNc; o += 8)
        Z += expf(fabsf(ss[row * SSTR + row + o]) - m);
      for (int mm = 1; mm < 8; mm <<= 1) Z += __shfl_xor(Z, mm, 32);
      const float inv = 1.0f / Z;
      for (int o = sub; o < WNc; o += 8) {
        const float s = ss[row * SSTR + row + o];
        const float p = expf(fabsf(s) - m) * inv;
        const float wv = (s > 0.0f) ? p : ((s < 0.0f) ? -p : 0.0f);
        wm[row * WSTR + row + o] = wv * sw[row + o];
      }
    } else if (g < Nc) {
      float m = -1e30f;
      for (int o = sub; o < WNc; o += 8) {
        const int gn = g - Wc + o;
        if (gn >= 0 && gn < Nc) m = fmaxf(m, fabsf(ss[row * SSTR + row + o]));
      }
      for (int mm = 1; mm < 8; mm <<= 1) m = fmaxf(m, __shfl_xor(m, mm, 32));
      float Z = 0.0f;
      for (int o = sub; o < WNc; o += 8) {
        const int gn = g - Wc + o;
        if (gn >= 0 && gn < Nc)
          Z += expf(fabsf(ss[row * SSTR + row + o]) - m);
      }
      for (int mm = 1; mm < 8; mm <<= 1) Z += __shfl_xor(Z, mm, 32);
      const float inv = 1.0f / Z;
      for (int o = sub; o < WNc; o += 8) {
        const int gn = g - Wc + o;
        if (gn >= 0 && gn < Nc) {
          const float s = ss[row * SSTR + row + o];
          const float p = expf(fabsf(s) - m) * inv;
          const float wv = (s > 0.0f) ? p : ((s < 0.0f) ? -p : 0.0f);
          wm[row * WSTR + row + o] = wv * sw[row + o];
        }
      }
    }
  }
  __syncthreads();

  // delta[16,512] = Wmat[16,144] * Xwin[144,512]; residual add into val.
  const float* xw = x + ((size_t)b * Nc + n0 - Wc) * Dc;  // window base row
  for (int ct = wave; ct < Dc / 16; ct += 4) {
    const int cb = ct * 16;
    v8f acc = {0.f, 0.f, 0.f, 0.f, 0.f, 0.f, 0.f, 0.f};
    if (interior) {
      for (int kk = 0; kk < WINc / 4; ++kk) {
        const int c0 = 4 * kk + 2 * hf;
        v2f a;
        a.x = wm[l16 * WSTR + c0];
        a.y = wm[l16 * WSTR + c0 + 1];
        v2f bv;
        bv.x = xw[(size_t)c0 * Dc + cb + l16];
        bv.y = xw[(size_t)(c0 + 1) * Dc + cb + l16];
        acc = wmma_f32(a, bv, acc);
      }
    } else {
      for (int kk = 0; kk < WINc / 4; ++kk) {
        const int c0 = 4 * kk + 2 * hf;
        v2f a;
        a.x = wm[l16 * WSTR + c0];
        a.y = wm[l16 * WSTR + c0 + 1];
        const int g0 = n0 - Wc + c0, g1 = g0 + 1;  // guards uniform per 16-lane half
        v2f bv;
        bv.x = (g0 >= 0 && g0 < Nc) ? x[((size_t)b * Nc + g0) * Dc + cb + l16]
                                    : 0.0f;
        bv.y = (g1 >= 0 && g1 < Nc) ? x[((size_t)b * Nc + g1) * Dc + cb + l16]
                                    : 0.0f;
        acc = wmma_f32(a, bv, acc);
      }
    }
    if (full) {
      float* vb = val + ((size_t)b * Nc + n0) * Dc + cb + l16;
      for (int i = 0; i < 8; ++i) {
        const int r = i + 8 * hf;
        vb[(size_t)r * Dc] += acc[i];
      }
    } else {
      for (int i = 0; i < 8; ++i) {
        const int r = i + 8 * hf, gg = n0 + r;
        if (gg < Nc) val[((size_t)b * Nc + gg) * Dc + cb + l16] += acc[i];
      }
    }
  }
}

// ---------------------------------------------------------------------------
extern "C" void kernel_launch(void* const* d_in, const int* in_sizes, int n_in,
                              void* d_out, int out_size, void* d_ws,
                              size_t ws_size, hipStream_t stream) {
  const int* ids = (const int*)d_in[0];
  const float* emb = (const float*)d_in[1];
  const float* pos = (const float*)d_in[2];
  const float* anchor_val = (const float*)d_in[3];
  const float* anchor_state = (const float*)d_in[4];
  const float* in_gamma = (const float*)d_in[5];
  const float* in_beta = (const float*)d_in[6];
  const float* ng = (const float*)d_in[7];
  const float* nb = (const float*)d_in[8];
  const float* Uw = (const float*)d_in[9];
  const float* Vw = (const float*)d_in[10];
  const float* sp_w = (const float*)d_in[11];
  const float* sp_b = (const float*)d_in[12];

  float* val = (float*)d_out;  // residual stream lives in d_out
  float* ws = (float*)d_ws;
  float* x = ws;                              // [B,N,D]
  float* q = x + (size_t)Bc * Nc * Dc;        // [B,N,R]
  float* k = q + (size_t)Bc * Nc * Rc;        // [B,N,R]
  float* st = k + (size_t)Bc * Nc * Rc;       // [B,N]

  embed_ln_kernel<<<Bc * Nc, 256, 0, stream>>>(
      ids, emb, pos, anchor_val, anchor_state, in_gamma, in_beta, sp_w, sp_b,
      val, st);

  for (int l = 0; l < Lc; ++l) {
    ln_proj_kernel<<<Bc * NT, 128, 0, stream>>>(
        val, ng + (size_t)l * Dc, nb + (size_t)l * Dc, Uw + (size_t)l * Dc * Rc,
        Vw + (size_t)l * Dc * Rc, x, q, k);
    attn_kernel<<<Bc * NT, 128, 0, stream>>>(x, q, k, st, val);
  }
}